// GCN_JK_EEG_54202487275903
// MI455X (gfx1250) — compile-verified
//
#include <hip/hip_runtime.h>
#include <math.h>

// ---- problem constants (match reference) ----
#define N_NODES   100000
#define N_EDGES   3200000
#define IN_DIM    128
#define HID       128
#define OUT_DIM   2
#define LAYERS    3
#define PART      64
#define GRAPHS    64
#define EPSBN     1e-5f
#define JKDIM     (HID * LAYERS)   // 384

typedef __attribute__((ext_vector_type(2))) float v2f;
typedef __attribute__((ext_vector_type(8))) float v8f;

// ------------------------------------------------------------------
// degree / normalization
// ------------------------------------------------------------------
__global__ void k_init_deg(float* deg) {
    int i = blockIdx.x * blockDim.x + threadIdx.x;
    if (i < N_NODES) deg[i] = 1.0f;               // self loop contributes 1
}

__global__ void k_accum_deg(const int* ei, float* deg) {
    int e = blockIdx.x * blockDim.x + threadIdx.x;
    if (e < N_EDGES) {
        int col = ei[N_EDGES + e];                // target node
        atomicAdd(&deg[col], 1.0f);
    }
}

__global__ void k_deg_to_dis(float* dis) {
    int i = blockIdx.x * blockDim.x + threadIdx.x;
    if (i < N_NODES) dis[i] = rsqrtf(dis[i]);     // deg >= 1 always
}

// ------------------------------------------------------------------
// init hjk layer slice with conv bias (scatter target)
// ------------------------------------------------------------------
__global__ void k_init_slice(float* dst /* hjk + l*HID */, const float* bias /* b_conv + l*HID */) {
    long long i = (long long)blockIdx.x * blockDim.x + threadIdx.x;
    if (i < (long long)N_NODES * HID) {
        long long n = i >> 7;                     // /128
        int c = (int)(i & 127);
        dst[n * JKDIM + c] = bias[c];
    }
}

// ------------------------------------------------------------------
// block-diagonal linear:  xw[:,p*64:(p+1)*64] = in[:,p*64:(p+1)*64] @ W[l,p]
// one wave per 16x16 output tile, V_WMMA_F32_16X16X4_F32, K = 64 in steps of 4.
// grid = (N/16) * 8 tiles, blockDim = 32 (full wave, EXEC all ones).
// ------------------------------------------------------------------
__global__ void __launch_bounds__(32)
k_gemm_wmma(const float* __restrict__ in, long long in_stride,
            const float* __restrict__ Wl /* W_conv + l*2*64*64 */,
            float* __restrict__ xw) {
    int tile = blockIdx.x;
    int mt = tile >> 3;                 // node tile (16 rows)
    int nt = tile & 7;                  // column tile (16 of 128 cols)
    int p  = nt >> 2;                   // which 64-wide part
    const float* Wp = Wl + (long long)p * PART * PART;

    int lane  = threadIdx.x;
    int mrow  = mt * 16 + (lane & 15);              // A row for this lane
    int khalf = lane >> 4;                          // 0: K pair {0,1}, 1: {2,3}
    int ncol  = nt * 16 + (lane & 15);              // global output column
    int wcol  = ncol - p * PART;                    // column within the part

    const float* arow = in + (long long)mrow * in_stride + p * PART;

    v8f acc = {};
    #pragma unroll
    for (int kk = 0; kk < PART; kk += 4) {
        int ka = kk + khalf * 2;
        v2f a, b;
        a.x = arow[ka];
        a.y = arow[ka + 1];
        b.x = Wp[(long long)ka * PART + wcol];
        b.y = Wp[(long long)(ka + 1) * PART + wcol];
        acc = __builtin_amdgcn_wmma_f32_16x16x4_f32(
            /*neg_a=*/false, a, /*neg_b=*/false, b,
            /*c_mod=*/(short)0, acc, /*reuse_a=*/false, /*reuse_b=*/false);
    }

    // C/D layout: VGPR v -> M = v (lanes 0-15) or v+8 (lanes 16-31); N = lane%16
    int mbase = mt * 16 + khalf * 8;
    int scol  = nt * 16 + (lane & 15);
    #pragma unroll
    for (int v = 0; v < 8; ++v)
        xw[(long long)(mbase + v) * HID + scol] = acc[v];
}

// ------------------------------------------------------------------
// edge gather-scale-scatter:  dst[col] += xw[row] * norm
// one wave per edge, each lane handles 4 consecutive features.
// edges e < E are real edges; e >= E are self loops (node e-E).
// ------------------------------------------------------------------
__global__ void k_edge_scatter(const int* __restrict__ ei, const float* __restrict__ dis,
                               const float* __restrict__ xw,
                               float* __restrict__ dst /* hjk + l*HID, row stride JKDIM */) {
    long long idx = (long long)blockIdx.x * blockDim.x + threadIdx.x;
    long long e = idx >> 5;
    int lane = (int)(idx & 31);
    if (e >= (long long)(N_EDGES + N_NODES)) return;

    int r, c; float nrm;
    if (e < N_EDGES) {
        r = ei[e];
        c = ei[N_EDGES + e];
        nrm = dis[r] * dis[c];
    } else {
        r = c = (int)(e - N_EDGES);
        float d = dis[r];
        nrm = d * d;
    }
    const float4 v = *(const float4*)(xw + (long long)r * HID + lane * 4);
    float* o = dst + (long long)c * JKDIM + lane * 4;
    atomicAdd(o + 0, v.x * nrm);
    atomicAdd(o + 1, v.y * nrm);
    atomicAdd(o + 2, v.z * nrm);
    atomicAdd(o + 3, v.w * nrm);
}

// ------------------------------------------------------------------
// eval-mode BatchNorm + ReLU, in place on one hjk slice
// ------------------------------------------------------------------
__global__ void k_bn_relu(float* __restrict__ dst /* hjk + l*HID */,
                          const float* __restrict__ gamma, const float* __restrict__ beta,
                          const float* __restrict__ mean,  const float* __restrict__ var) {
    long long i = (long long)blockIdx.x * blockDim.x + threadIdx.x;
    if (i < (long long)N_NODES * HID) {
        long long n = i >> 7;
        int c = (int)(i & 127);
        float* p = dst + n * JKDIM + c;
        float v = (*p - mean[c]) * rsqrtf(var[c] + EPSBN) * gamma[c] + beta[c];
        *p = fmaxf(v, 0.0f);
    }
}

// ------------------------------------------------------------------
// global mean pool
// ------------------------------------------------------------------
__global__ void k_zero_pool(float* pooled, float* cnt) {
    int i = blockIdx.x * blockDim.x + threadIdx.x;
    if (i < GRAPHS * JKDIM) pooled[i] = 0.0f;
    if (i < GRAPHS) cnt[i] = 0.0f;
}

__global__ void k_pool_accum(const int* __restrict__ batch, const float* __restrict__ hjk,
                             float* __restrict__ pooled) {
    long long i = (long long)blockIdx.x * blockDim.x + threadIdx.x;
    if (i < (long long)N_NODES * JKDIM) {
        long long n = i / JKDIM;
        int c = (int)(i - n * JKDIM);
        int g = batch[n];
        atomicAdd(&pooled[(long long)g * JKDIM + c], hjk[i]);
    }
}

__global__ void k_cnt_accum(const int* __restrict__ batch, float* __restrict__ cnt) {
    int n = blockIdx.x * blockDim.x + threadIdx.x;
    if (n < N_NODES) atomicAdd(&cnt[batch[n]], 1.0f);
}

__global__ void k_pool_div(float* pooled, const float* cnt) {
    int i = blockIdx.x * blockDim.x + threadIdx.x;
    if (i < GRAPHS * JKDIM) {
        int g = i / JKDIM;
        pooled[i] /= fmaxf(cnt[g], 1.0f);
    }
}

// ------------------------------------------------------------------
// FC1 + ReLU + FC2 + log_softmax  (one block per graph, 128 threads)
// ------------------------------------------------------------------
__global__ void __launch_bounds__(128)
k_fc(const float* __restrict__ pooled,
     const float* __restrict__ W1, const float* __restrict__ b1,
     const float* __restrict__ W2, const float* __restrict__ b2,
     float* __restrict__ out) {
    __shared__ float z1[HID];
    __shared__ float z2[OUT_DIM];
    int g = blockIdx.x, t = threadIdx.x;
    const float* pr = pooled + (long long)g * JKDIM;

    float s = b1[t];
    for (int k = 0; k < JKDIM; ++k)
        s = fmaf(pr[k], W1[(long long)k * HID + t], s);
    z1[t] = fmaxf(s, 0.0f);
    __syncthreads();

    if (t < OUT_DIM) {
        float s2 = b2[t];
        for (int k = 0; k < HID; ++k)
            s2 = fmaf(z1[k], W2[k * OUT_DIM + t], s2);
        z2[t] = s2;
    }
    __syncthreads();

    if (t < OUT_DIM) {
        float m = fmaxf(z2[0], z2[1]);
        float lse = m + logf(expf(z2[0] - m) + expf(z2[1] - m));
        out[g * OUT_DIM + t] = z2[t] - lse;
    }
}

// ------------------------------------------------------------------
extern "C" void kernel_launch(void* const* d_in, const int* in_sizes, int n_in,
                              void* d_out, int out_size, void* d_ws, size_t ws_size,
                              hipStream_t stream) {
    const float* x        = (const float*)d_in[0];
    const int*   ei       = (const int*)  d_in[1];
    const int*   batch    = (const int*)  d_in[2];
    const float* W_conv   = (const float*)d_in[3];   // (L,P,64,64)
    const float* b_conv   = (const float*)d_in[4];   // (L,128)
    const float* bn_gamma = (const float*)d_in[5];
    const float* bn_beta  = (const float*)d_in[6];
    const float* bn_mean  = (const float*)d_in[7];
    const float* bn_var   = (const float*)d_in[8];
    const float* W_fc1    = (const float*)d_in[9];   // (384,128)
    const float* b_fc1    = (const float*)d_in[10];
    const float* W_fc2    = (const float*)d_in[11];  // (128,2)
    const float* b_fc2    = (const float*)d_in[12];
    float* out = (float*)d_out;

    // workspace layout (floats)
    float* ws     = (float*)d_ws;
    float* xw     = ws;                                        // N*128
    float* hjk    = xw  + (long long)N_NODES * HID;            // N*384
    float* dis    = hjk + (long long)N_NODES * JKDIM;          // N
    float* pooled = dis + N_NODES;                             // 64*384
    float* cnt    = pooled + GRAPHS * JKDIM;                   // 64

    const int T = 256;
    // ---- normalization coefficients ----
    k_init_deg  <<<(N_NODES + T - 1) / T, T, 0, stream>>>(dis);
    k_accum_deg <<<(N_EDGES + T - 1) / T, T, 0, stream>>>(ei, dis);
    k_deg_to_dis<<<(N_NODES + T - 1) / T, T, 0, stream>>>(dis);

    const long long nh  = (long long)N_NODES * HID;
    const long long nsc = (long long)(N_EDGES + N_NODES) * 32;

    for (int l = 0; l < LAYERS; ++l) {
        const float* in_ptr   = (l == 0) ? x : (hjk + (long long)(l - 1) * HID);
        long long    in_stride = (l == 0) ? IN_DIM : JKDIM;
        float*       slice    = hjk + (long long)l * HID;

        k_init_slice<<<(int)((nh + T - 1) / T), T, 0, stream>>>(slice, b_conv + l * HID);
        k_gemm_wmma <<<(N_NODES / 16) * 8, 32, 0, stream>>>(
            in_ptr, in_stride, W_conv + (long long)l * 2 * PART * PART, xw);
        k_edge_scatter<<<(int)((nsc + T - 1) / T), T, 0, stream>>>(ei, dis, xw, slice);
        k_bn_relu   <<<(int)((nh + T - 1) / T), T, 0, stream>>>(
            slice, bn_gamma + l * HID, bn_beta + l * HID, bn_mean + l * HID, bn_var + l * HID);
    }

    // ---- global mean pool ----
    k_zero_pool <<<(GRAPHS * JKDIM + T - 1) / T, T, 0, stream>>>(pooled, cnt);
    const long long np = (long long)N_NODES * JKDIM;
    k_pool_accum<<<(int)((np + T - 1) / T), T, 0, stream>>>(batch, hjk, pooled);
    k_cnt_accum <<<(N_NODES + T - 1) / T, T, 0, stream>>>(batch, cnt);
    k_pool_div  <<<(GRAPHS * JKDIM + T - 1) / T, T, 0, stream>>>(pooled, cnt);

    // ---- classifier head ----
    k_fc<<<GRAPHS, HID, 0, stream>>>(pooled, W_fc1, b_fc1, W_fc2, b_fc2, out);
}